// LightGCN_33105607918228
// MI455X (gfx1250) — compile-verified
//
#include <hip/hip_runtime.h>
#include <math.h>

#define FACTORS   128
#define N_USERS_C 100000
#define M_ITEMS_C 50000
#define N_NODES_C (N_USERS_C + M_ITEMS_C)
#define N_QUERY_C 100000

typedef __attribute__((ext_vector_type(2))) float v2f;
typedef __attribute__((ext_vector_type(8))) float v8f;

// ---------------------------------------------------------------------------
// init: concat [user_emb; item_emb] into cur AND acc (acc starts at layer-0 emb)
// ---------------------------------------------------------------------------
__global__ void lg_init_concat(const float4* __restrict__ uemb,
                               const float4* __restrict__ iemb,
                               float4* __restrict__ cur,
                               float4* __restrict__ acc) {
  long long i = (long long)blockIdx.x * blockDim.x + threadIdx.x;
  const long long uN = (long long)N_USERS_C * (FACTORS / 4);
  const long long tN = (long long)N_NODES_C * (FACTORS / 4);
  if (i >= tN) return;
  float4 v = (i < uN) ? uemb[i] : iemb[i - uN];
  cur[i] = v;
  acc[i] = v;
}

__global__ void lg_zero(float4* __restrict__ p) {
  long long i = (long long)blockIdx.x * blockDim.x + threadIdx.x;
  const long long tN = (long long)N_NODES_C * (FACTORS / 4);
  if (i >= tN) return;
  p[i] = make_float4(0.f, 0.f, 0.f, 0.f);
}

__global__ void lg_acc_add(float4* __restrict__ acc, const float4* __restrict__ x) {
  long long i = (long long)blockIdx.x * blockDim.x + threadIdx.x;
  const long long tN = (long long)N_NODES_C * (FACTORS / 4);
  if (i >= tN) return;
  float4 a = acc[i], b = x[i];
  a.x += b.x; a.y += b.y; a.z += b.z; a.w += b.w;
  acc[i] = a;
}

// ---------------------------------------------------------------------------
// SpMM scatter: one wave per edge; lane L handles factors [4L, 4L+3].
// Gather cur[col] (coalesced 512B per wave, hits L2 since table is 76.8MB),
// scale by edge_val, atomic-add into nxt[row].
// ---------------------------------------------------------------------------
__global__ void lg_scatter(const int* __restrict__ er,
                           const int* __restrict__ ec,
                           const float* __restrict__ ev,
                           const float* __restrict__ cur,
                           float* __restrict__ nxt,
                           int E) {
  long long tid = (long long)blockIdx.x * blockDim.x + threadIdx.x;
  int e = (int)(tid >> 5);
  if (e >= E) return;
  int c = ((int)tid & 31) * 4;
  int row = er[e];
  int col = ec[e];
  float val = ev[e];
  const float4 s = *(const float4*)(cur + (size_t)col * FACTORS + c);
  float* d = nxt + (size_t)row * FACTORS + c;
  atomicAdd(d + 0, val * s.x);
  atomicAdd(d + 1, val * s.y);
  atomicAdd(d + 2, val * s.z);
  atomicAdd(d + 3, val * s.w);
}

// ---------------------------------------------------------------------------
// Scoring with V_WMMA_F32_16X16X4_F32: one wave handles 16 queries.
// A = U-tile (16 x 4), B = I-tile^T (4 x 16); D accumulates U * I^T over
// K = 128 in 32 steps; diag(D) are the per-query dot products.
// Lane mapping (per documented 32-bit A layout: VGPR0 = K{0|2}, VGPR1 = K{1|3}):
//   lane L: query q0 + (L&15); K sub-offset = (L>>4)*2 for both A and B.
// final = acc/4 on both sides -> scale dot by 1/16 before sigmoid.
// ---------------------------------------------------------------------------
__global__ void lg_score_wmma(const float* __restrict__ acc,
                              const int* __restrict__ users,
                              const int* __restrict__ items,
                              float* __restrict__ out) {
  int tile = blockIdx.x * 2 + (threadIdx.x >> 5);   // 6250 tiles of 16 queries
  int lane = threadIdx.x & 31;
  int q0   = tile * 16;
  int q    = q0 + (lane & 15);
  int koff = (lane >> 4) * 2;

  const float* up = acc + (size_t)users[q] * FACTORS;
  const float* ip = acc + ((size_t)N_USERS_C + (size_t)items[q]) * FACTORS;

  v8f c = {0.f, 0.f, 0.f, 0.f, 0.f, 0.f, 0.f, 0.f};
#pragma unroll
  for (int k = 0; k < FACTORS; k += 4) {
    v2f a, b;
    a.x = up[k + koff];
    a.y = up[k + koff + 1];
    b.x = ip[k + koff];
    b.y = ip[k + koff + 1];
    c = __builtin_amdgcn_wmma_f32_16x16x4_f32(false, a, false, b,
                                              (short)0, c, false, false);
  }

  // Diagonal extraction: m in 0..7 -> lane m, c[m]; m in 8..15 -> lane m+16, c[m-8]
  bool has = (lane < 8) || (lane >= 24);
  int  vi  = (lane < 8) ? lane : (lane - 24);
  float dot = 0.f;
#pragma unroll
  for (int j = 0; j < 8; ++j) dot = (j == vi) ? c[j] : dot;

  if (has) {
    int m = (lane < 8) ? lane : (lane - 16);
    float s = dot * 0.0625f;                 // (acc/4)·(acc/4) = acc·acc / 16
    out[q0 + m] = 1.0f / (1.0f + expf(-s));
  }
}

// ---------------------------------------------------------------------------
extern "C" void kernel_launch(void* const* d_in, const int* in_sizes, int n_in,
                              void* d_out, int out_size, void* d_ws, size_t ws_size,
                              hipStream_t stream) {
  const float* user_emb = (const float*)d_in[0];
  const float* item_emb = (const float*)d_in[1];
  const int*   edge_row = (const int*)d_in[2];
  const int*   edge_col = (const int*)d_in[3];
  const float* edge_val = (const float*)d_in[4];
  const int*   users    = (const int*)d_in[5];
  const int*   items    = (const int*)d_in[6];
  float*       out      = (float*)d_out;

  const int E = in_sizes[2];                       // 4,000,000 edges

  const size_t NB = (size_t)N_NODES_C * FACTORS;   // floats per node buffer
  float* bufA = (float*)d_ws;
  float* bufB = bufA + NB;
  float* accB = bufB + NB;                         // needs ~230.4 MB of ws

  const int TB = 256;
  const long long nV4 = (long long)N_NODES_C * (FACTORS / 4);
  const int gV4 = (int)((nV4 + TB - 1) / TB);

  lg_init_concat<<<gV4, TB, 0, stream>>>((const float4*)user_emb,
                                         (const float4*)item_emb,
                                         (float4*)bufA, (float4*)accB);

  const float* cur = bufA;
  float*       nxt = bufB;
  const long long scatThreads = (long long)E * 32;
  const int gScat = (int)((scatThreads + TB - 1) / TB);

  for (int layer = 0; layer < 3; ++layer) {
    lg_zero<<<gV4, TB, 0, stream>>>((float4*)nxt);
    lg_scatter<<<gScat, TB, 0, stream>>>(edge_row, edge_col, edge_val, cur, nxt, E);
    lg_acc_add<<<gV4, TB, 0, stream>>>((float4*)accB, (const float4*)nxt);
    // ping-pong; previous input buffer is dead after this layer
    const float* t = cur;
    cur = nxt;
    nxt = (float*)t;
  }

  // 100000 queries = 6250 tiles of 16; 2 waves per block -> 3125 blocks
  lg_score_wmma<<<N_QUERY_C / 32, 64, 0, stream>>>(accB, users, items, out);
}